// FaceKernelCorrelation_49538152792829
// MI455X (gfx1250) — compile-verified
//
#include <hip/hip_runtime.h>
#include <hip/hip_bf16.h>
#include <math.h>

typedef __attribute__((ext_vector_type(2))) float v2f;
typedef __attribute__((ext_vector_type(8))) float v8f;

#define B_ 8
#define N_ 16384
#define K_ 64
#define EXP_SCALE (-12.5f)   /* -1/(2*0.2^2) */
#define LOG2E     1.4426950408889634f
#define BN_EPS    1e-5f

// Raw v_exp_f32 (args here are in [-72.2, 0] -> always normal, no fixup needed)
#if __has_builtin(__builtin_amdgcn_exp2f)
#define EXP2(x) __builtin_amdgcn_exp2f(x)
#else
#define EXP2(x) __expf((x) * 0.6931471805599453f)
#endif

// xor-lane add within wave32 via ds_swizzle (group-of-32: and=0x1f, xor=imm)
__device__ __forceinline__ float xor1_add(float v) {
    int s = __builtin_amdgcn_ds_swizzle(__float_as_int(v), 0x041F); // SWAPX1
    return v + __int_as_float(s);
}
__device__ __forceinline__ float xor2_add(float v) {
    int s = __builtin_amdgcn_ds_swizzle(__float_as_int(v), 0x081F); // SWAPX2
    return v + __int_as_float(s);
}

// ---------------------------------------------------------------------------
// Kernel 1: feat[b,k,n] = (1/16) * sum_{p,m} exp(-dist/(2 sigma^2))
// One wave handles 4 consecutive n (16 A-rows), all 64 k via 16 WMMA tiles.
// Dot products via V_WMMA_F32_16X16X4_F32 (contraction over xyz padded to 4).
// exp(dist*c) folded to exp2(KF*(f2+w2) + KD*dot): 1 FMA + 1 v_exp per term.
// ---------------------------------------------------------------------------
__global__ __launch_bounds__(256) void fkc_feat_kernel(
    const float* __restrict__ normals,   // (B,3,N)
    const int*   __restrict__ nbr,       // (B,N,3)
    const float* __restrict__ walpha,    // (K,4)
    const float* __restrict__ wbeta,     // (K,4)
    float* __restrict__ feat)            // (B,K,N)  == d_out
{
    __shared__ float sWx[256], sWy[256], sWz[256], sW2[256];
    __shared__ float sF2[8][16];         // per-wave ||f||^2 * KF, per A-row

    const int tid = threadIdx.x;
    const float KF = EXP_SCALE * LOG2E;  // coeff of (f2 + w2)
    const float KD = -2.0f * KF;         // coeff of the dot product

    // Build the 256 kernel-point directions (col = k*4 + m) once per block.
    {
        float a  = walpha[tid];
        float bt = wbeta[tid];
        float sa = sinf(a),  ca = cosf(a);
        float sb = sinf(bt), cb = cosf(bt);
        float wx = sa * cb, wy = sa * sb, wz = ca;
        sWx[tid] = wx; sWy[tid] = wy; sWz[tid] = wz;
        sW2[tid] = (wx * wx + wy * wy + wz * wz) * KF;
    }
    __syncthreads();

    const int lane = tid & 31;
    const int wave = tid >> 5;
    const int b    = blockIdx.x >> 9;     // 512 blocks per batch (16384/32)
    const int nb   = blockIdx.x & 511;
    const int n_base = nb * 32 + wave * 4;

    // ---- Load A-matrix (16 rows x 4): row r = (n_local, p), K = xyz + pad ---
    const int r = lane & 15;
    const int n = n_base + (r >> 2);
    const int p = r & 3;
    const int idx = (p == 0) ? n : nbr[(b * N_ + n) * 3 + (p - 1)];

    const float* nrm = normals + (size_t)b * 3 * N_;
    const float x = nrm[idx];
    const float y = nrm[N_ + idx];
    const float z = nrm[2 * N_ + idx];

    // Stage pre-scaled ||f||^2 per row (lanes r and r+16 write the same value).
    sF2[wave][r] = (x * x + y * y + z * z) * KF;

    const bool lowh = (lane < 16);
    v2f av;                               // lanes 0-15: (x,y) ; 16-31: (z,0)
    av.x = lowh ? x : z;
    av.y = lowh ? y : 0.0f;

    __syncthreads();

    // Each lane accumulates D rows j+8*half: pull their f2*KF into registers.
    float f2l[8];
    const int rowoff = lowh ? 0 : 8;
#pragma unroll
    for (int j = 0; j < 8; ++j)
        f2l[j] = sF2[wave][rowoff + j];

    // Per-lane constant selects for the B-matrix halves.
    const float* __restrict__ pBx = lowh ? sWx : sWz;  // c=0 / c=2 row
    const float ymask = lowh ? 1.0f : 0.0f;            // c=1 / zero-pad row

    float* featb = feat + (size_t)b * K_ * N_;

#pragma unroll
    for (int t = 0; t < 16; ++t) {
        const int col = t * 16 + (lane & 15);   // (k,m): k=col>>2, m=col&3
        v2f bv;
        bv.x = pBx[col];
        bv.y = sWy[col] * ymask;
        const float base = sW2[col];            // already * KF

        v8f acc = {};
        acc = __builtin_amdgcn_wmma_f32_16x16x4_f32(
            /*neg_a=*/false, av, /*neg_b=*/false, bv,
            /*c_mod=*/(short)0, acc, /*reuse_a=*/false, /*reuse_b=*/false);

        float s0 = 0.0f, s1 = 0.0f;             // n_local {0,2} and {1,3}
#pragma unroll
        for (int j = 0; j < 8; ++j) {
            float e = EXP2(fmaf(acc[j], KD, f2l[j] + base));
            if (j < 4) s0 += e; else s1 += e;   // sums over face points p
        }
        // Sum over kernel points m: 4 adjacent lanes share the same k.
        s0 = xor2_add(xor1_add(s0));
        s1 = xor2_add(xor1_add(s1));

        if ((lane & 3) == 0) {
            const int k   = t * 4 + ((lane & 15) >> 2);
            const int nlo = n_base + (lowh ? 0 : 2);
            featb[(size_t)k * N_ + nlo]     = s0 * 0.0625f;
            featb[(size_t)k * N_ + nlo + 1] = s1 * 0.0625f;
        }
    }
}

// ---------------------------------------------------------------------------
// Kernel 2: per-channel BN statistics -> scale/shift (deterministic tree sum)
// ---------------------------------------------------------------------------
__global__ __launch_bounds__(256) void fkc_stats_kernel(
    const float* __restrict__ feat,
    const float* __restrict__ gamma,
    const float* __restrict__ beta,
    float* __restrict__ stats)           // [0,64): scale  [64,128): shift
{
    __shared__ float ssum[256], ssum2[256];
    const int k = blockIdx.x;
    const int tid = threadIdx.x;
    float s = 0.0f, s2 = 0.0f;
    for (int b = 0; b < B_; ++b) {
        const float* f = feat + ((size_t)b * K_ + k) * N_;
        for (int i = tid; i < N_; i += 256) {
            float v = f[i];
            s += v; s2 += v * v;
        }
    }
    ssum[tid] = s; ssum2[tid] = s2;
    __syncthreads();
    for (int off = 128; off > 0; off >>= 1) {
        if (tid < off) {
            ssum[tid]  += ssum[tid + off];
            ssum2[tid] += ssum2[tid + off];
        }
        __syncthreads();
    }
    if (tid == 0) {
        const float cnt = (float)(B_ * N_);
        float mean = ssum[0] / cnt;
        float var  = ssum2[0] / cnt - mean * mean;   // biased, matches jnp.var
        float sc   = gamma[k] * rsqrtf(var + BN_EPS);
        stats[k]      = sc;
        stats[K_ + k] = beta[k] - mean * sc;
    }
}

// ---------------------------------------------------------------------------
// Kernel 3: in-place affine + ReLU
// ---------------------------------------------------------------------------
__global__ __launch_bounds__(256) void fkc_bn_relu_kernel(
    float* __restrict__ out, const float* __restrict__ stats)
{
    const size_t idx = (size_t)blockIdx.x * 256 + threadIdx.x;
    const int k = (int)((idx >> 14) & (K_ - 1));
    float y = stats[k] * out[idx] + stats[K_ + k];
    out[idx] = fmaxf(y, 0.0f);
}

extern "C" void kernel_launch(void* const* d_in, const int* in_sizes, int n_in,
                              void* d_out, int out_size, void* d_ws, size_t ws_size,
                              hipStream_t stream) {
    const float* normals = (const float*)d_in[0];
    const int*   nbr     = (const int*)  d_in[1];
    const float* walpha  = (const float*)d_in[2];
    const float* wbeta   = (const float*)d_in[3];
    const float* gamma   = (const float*)d_in[4];
    const float* beta    = (const float*)d_in[5];
    float* out   = (float*)d_out;
    float* stats = (float*)d_ws;

    // 8 batches * 512 n-chunks of 32 -> 4096 blocks; wave handles 4 n.
    fkc_feat_kernel<<<4096, 256, 0, stream>>>(normals, nbr, walpha, wbeta, out);
    fkc_stats_kernel<<<K_, 256, 0, stream>>>(out, gamma, beta, stats);
    fkc_bn_relu_kernel<<<(B_ * K_ * N_) / 256, 256, 0, stream>>>(out, stats);
}